// MDEformer_29583734735561
// MI455X (gfx1250) — compile-verified
//
#include <hip/hip_runtime.h>
#include <math.h>

// ---------------------------------------------------------------------------
// MDEformer block for gfx1250 (CDNA5, wave32, WMMA).
// All dense GEMMs: v_wmma_f32_16x16x4_f32 with guard-free inner loops —
// padded buffers + pre-transposed weights => every fragment is one b64 load.
// FFN stages its A-tile via async global->LDS (when available) and prefetches
// streamed weight rows. Workspace: ~53.1M floats (~213 MB).
// ---------------------------------------------------------------------------

#define NN   170
#define SEQT 12
#define BB   32
#define DM   120
#define NH   8
#define HD   15
#define FF   2048
#define BT   (BB*SEQT)        // 384
#define ROWS (BT*NN)          // 65280
#define NNP  176              // 11*16 padded nodes
#define NNP2 (NNP*NNP)        // 30976
#define DMP  128              // padded channels
#define NTI  11
#define LEPS 1e-5f
#define PBSZ ((size_t)BT*NNP*DMP)   // padded activation plane: 8,650,752 floats

typedef __attribute__((ext_vector_type(2))) float v2f;
typedef __attribute__((ext_vector_type(8))) float v8f;
typedef int v4i __attribute__((vector_size(4 * sizeof(int))));

#if defined(__has_builtin)
#if __has_builtin(__builtin_amdgcn_global_load_async_to_lds_b128) && \
    __has_builtin(__builtin_amdgcn_s_wait_asynccnt)
#define HAVE_ASYNC_LDS 1
#endif
#endif

__device__ __forceinline__ v8f wmma4(v2f a, v2f b, v8f c) {
  return __builtin_amdgcn_wmma_f32_16x16x4_f32(false, a, false, b, (short)0, c,
                                               false, false);
}
__device__ __forceinline__ v2f ld2(const float* p) { return *(const v2f*)p; }

__device__ __forceinline__ float wave_sum(float s) {
  for (int off = 16; off > 0; off >>= 1) s += __shfl_xor(s, off, 32);
  return s;
}

// ---------------------------------------------------------------------------
// P0: transposed+padded weight copy: out[mat][n*KP+k] = in[mat][k*N+n] (0 pad)
// ---------------------------------------------------------------------------
__global__ __launch_bounds__(256)
void k_transpose_pad(const float* __restrict__ in, float* __restrict__ out,
                     int K, int N, int KP, int NP) {
  const size_t mat = blockIdx.y;
  const int idx = blockIdx.x * 256 + threadIdx.x;
  if (idx >= NP * KP) return;
  const int n = idx / KP, k = idx % KP;
  const float v = in[mat * (size_t)K * N + (size_t)min(k, K - 1) * N + min(n, N - 1)];
  out[mat * (size_t)NP * KP + idx] = (n < N && k < K) ? v : 0.f;
}

// P1: xt[b2][c][node] = x[b2][node][c], zero padded to (DMP, NNP)
__global__ __launch_bounds__(256)
void k_build_xt(const float* __restrict__ x, float* __restrict__ xt) {
  const size_t idx = (size_t)blockIdx.x * 256 + threadIdx.x;
  const int nd = (int)(idx % NNP);
  const int c = (int)((idx / NNP) % DMP);
  const size_t b2 = idx / ((size_t)NNP * DMP);
  const float v = x[(b2 * NN + min(nd, NN - 1)) * DM + min(c, DM - 1)];
  xt[idx] = (c < DM && nd < NN) ? v : 0.f;
}

// P2: 176x176 transpose (for L -> Lt)
__global__ __launch_bounds__(256)
void k_transpose_g(const float* __restrict__ in, float* __restrict__ out) {
  const int br = blockIdx.y;
  const int idx = blockIdx.x * 256 + threadIdx.x;
  const int row = idx / NNP, col = idx % NNP;
  out[(size_t)br * NNP2 + (size_t)col * NNP + row] = in[(size_t)br * NNP2 + idx];
}

// ---------------------------------------------------------------------------
// K1: a = gate*relu(w1 w2^T - w2 w1^T + diag(beta)) + (1-gate)*adj  (padded)
// ---------------------------------------------------------------------------
__global__ __launch_bounds__(32)
void k_graph_new(const float* __restrict__ adj, const float* __restrict__ w1,
                 const float* __restrict__ w2, const float* __restrict__ beta,
                 const float* __restrict__ cw, const float* __restrict__ cb,
                 float* __restrict__ outA) {
  const int br = blockIdx.y;
  const int tm = blockIdx.x / NTI, tn = blockIdx.x % NTI;
  const int lane = threadIdx.x & 31, ln = lane & 15, hi = lane >> 4;
  const int mr = min(tm * 16 + ln, NN - 1);
  const int nr = min(tn * 16 + ln, NN - 1);
  const float* W1 = w1 + (size_t)br * NN * DM;
  const float* W2 = w2 + (size_t)br * NN * DM;
  const float* W1m = W1 + (size_t)mr * DM;
  const float* W2m = W2 + (size_t)mr * DM;
  const float* W1n = W1 + (size_t)nr * DM;
  const float* W2n = W2 + (size_t)nr * DM;
  v8f acc1 = {}, acc2 = {};
  for (int k0 = 0; k0 < DM; k0 += 4) {
    const int ka = k0 + 2 * hi;
    acc1 = wmma4(ld2(W1m + ka), ld2(W2n + ka), acc1);   // W1 @ W2^T
    acc2 = wmma4(ld2(W2m + ka), ld2(W1n + ka), acc2);   // W2 @ W1^T
  }
  const float c0 = cw[br * 2 + 0], c1 = cw[br * 2 + 1], c2 = cb[br];
  float* O = outA + (size_t)br * NNP2;
  for (int r = 0; r < 8; ++r) {
    const int row = tm * 16 + r + 8 * hi, col = tn * 16 + ln;
    const int rc = min(row, NN - 1), cc = min(col, NN - 1);
    float nv = acc1[r] - acc2[r] + ((row == col) ? beta[br * NN + rc] : 0.f);
    nv = fmaxf(nv, 0.f);
    const float ad = adj[rc * NN + cc];
    const float gate = 1.f / (1.f + expf(-(c0 * nv + c1 * ad + c2)));
    const float val = gate * nv + (1.f - gate) * ad;
    O[(size_t)row * NNP + col] = (row < NN && col < NN) ? val : 0.f;
  }
}

// ---------------------------------------------------------------------------
// K2: d[br][row] = rsqrt(rowsum) for valid rows, 0 for pad rows
// ---------------------------------------------------------------------------
__global__ __launch_bounds__(32)
void k_rowsum_rsqrt(const float* __restrict__ a, float* __restrict__ d) {
  const int br = blockIdx.y, row = blockIdx.x, lane = threadIdx.x;
  const float* p = a + (size_t)br * NNP2 + (size_t)row * NNP;
  float s = 0.f;
  for (int c = lane; c < NNP; c += 32) s += p[c];
  s = wave_sum(s);
  if (lane == 0) d[br * 256 + row] = (row < NN) ? rsqrtf(s) : 0.f;
}

// ---------------------------------------------------------------------------
// K3: mode 0: relu(d_r*in*d_c - eps); mode 1: d_r*in*d_c; mode 2: I - d_r*in*d_c
// ---------------------------------------------------------------------------
__global__ __launch_bounds__(256)
void k_sym_norm(const float* __restrict__ in, const float* __restrict__ d,
                float* __restrict__ out, int mode, float eps) {
  const int br = blockIdx.y;
  const int idx = blockIdx.x * 256 + threadIdx.x;
  const int row = idx / NNP, col = idx % NNP;
  float v = d[br * 256 + row] * in[(size_t)br * NNP2 + idx] * d[br * 256 + col];
  if (mode == 0) v = fmaxf(v - eps, 0.f);
  else if (mode == 2) v = ((row == col) ? 1.f : 0.f) - v;
  out[(size_t)br * NNP2 + idx] = (row < NN && col < NN) ? v : 0.f;
}

// ---------------------------------------------------------------------------
// K4: T2 = 2*L@L - I (uses L and Lt; fully padded, guard-free loop)
// ---------------------------------------------------------------------------
__global__ __launch_bounds__(32)
void k_cheb_t2(const float* __restrict__ L, const float* __restrict__ Lt,
               float* __restrict__ T2) {
  const int br = blockIdx.y;
  const int tm = blockIdx.x / NTI, tn = blockIdx.x % NTI;
  const int lane = threadIdx.x & 31, ln = lane & 15, hi = lane >> 4;
  const float* Arow = L + (size_t)br * NNP2 + (size_t)(tm * 16 + ln) * NNP;
  const float* Brow = Lt + (size_t)br * NNP2 + (size_t)(tn * 16 + ln) * NNP;
  v8f acc = {};
  for (int k0 = 0; k0 < NNP; k0 += 4) {
    const int ka = k0 + 2 * hi;
    acc = wmma4(ld2(Arow + ka), ld2(Brow + ka), acc);
  }
  float* O = T2 + (size_t)br * NNP2;
  for (int r = 0; r < 8; ++r) {
    const int row = tm * 16 + r + 8 * hi, col = tn * 16 + ln;
    const float val = 2.f * acc[r] - ((row == col) ? 1.f : 0.f);
    O[(size_t)row * NNP + col] = (row < NN && col < NN) ? val : 0.f;
  }
}

// ---------------------------------------------------------------------------
// K5: Z[which][b2] = (which ? T2 : L) @ X[b2]; B from xt (transposed); 1x2 block
// ---------------------------------------------------------------------------
__global__ __launch_bounds__(256)
void k_cheb_z(const float* __restrict__ L, const float* __restrict__ T2,
              const float* __restrict__ xt, float* __restrict__ Z, int br) {
  const int gid = blockIdx.x * 8 + (threadIdx.x >> 5);
  const int b2 = gid / 88;
  const int rem = gid % 88;
  const int which = rem / 44;
  const int t = rem % 44;
  const int tm = t / 4, tnp = t % 4;
  if (b2 >= BT) return;
  const int lane = threadIdx.x & 31, ln = lane & 15, hi = lane >> 4;
  const float* Arow = (which ? T2 : L) + (size_t)br * NNP2 + (size_t)(tm * 16 + ln) * NNP;
  const float* B0 = xt + ((size_t)b2 * DMP + tnp * 32 + ln) * NNP;
  const float* B1 = B0 + (size_t)16 * NNP;
  v8f acc0 = {}, acc1 = {};
  for (int k0 = 0; k0 < NNP; k0 += 4) {
    const int ka = k0 + 2 * hi;
    const v2f a = ld2(Arow + ka);
    acc0 = wmma4(a, ld2(B0 + ka), acc0);
    acc1 = wmma4(a, ld2(B1 + ka), acc1);
  }
  float* O = Z + (size_t)which * PBSZ + (size_t)b2 * NNP * DMP;
  for (int r = 0; r < 8; ++r) {
    const int row = tm * 16 + r + 8 * hi;
    O[(size_t)row * DMP + tnp * 32 + ln] = acc0[r];
    O[(size_t)row * DMP + tnp * 32 + 16 + ln] = acc1[r];
  }
}

// ---------------------------------------------------------------------------
// K6: Y = relu(X@W0 + Z1@W1 + Z2@W2 + bias), K fused over 3 segments; 1x2 block
// ---------------------------------------------------------------------------
__global__ __launch_bounds__(256)
void k_cheb_y(const float* __restrict__ x, const float* __restrict__ Z,
              const float* __restrict__ cwt, const float* __restrict__ bias,
              float* __restrict__ out, int br) {
  const int gid = blockIdx.x * 8 + (threadIdx.x >> 5);
  const int b2 = gid / 44;
  const int t = gid % 44;
  const int tm = t / 4, tnp = t % 4;
  if (b2 >= BT) return;
  const int lane = threadIdx.x & 31, ln = lane & 15, hi = lane >> 4;
  const int mrow = tm * 16 + ln;
  const float* A0 = x + ((size_t)b2 * NN + min(mrow, NN - 1)) * DM;   // dense x
  const float* A1 = Z + ((size_t)b2 * NNP + mrow) * DMP;              // Z1 padded
  const float* A2 = A1 + PBSZ;                                        // Z2
  const int n0 = tnp * 32 + ln;
  v8f acc0 = {}, acc1 = {};
  for (int seg = 0; seg < 3; ++seg) {
    const float* Arow = (seg == 0) ? A0 : (seg == 1) ? A1 : A2;
    const float* W = cwt + (size_t)(br * 3 + seg) * DMP * DMP;
    const float* B0 = W + (size_t)n0 * DMP;
    const float* B1 = B0 + (size_t)16 * DMP;
    for (int k0 = 0; k0 < DM; k0 += 4) {
      const int ka = k0 + 2 * hi;
      const v2f a = ld2(Arow + ka);
      acc0 = wmma4(a, ld2(B0 + ka), acc0);
      acc1 = wmma4(a, ld2(B1 + ka), acc1);
    }
  }
  float* O = out + (size_t)b2 * NNP * DMP;
  const float* bb = bias + br * DM;
  for (int r = 0; r < 8; ++r) {
    const int row = tm * 16 + r + 8 * hi;
    const int c0 = tnp * 32 + ln, c1 = c0 + 16;
    O[(size_t)row * DMP + c0] = fmaxf(acc0[r] + bb[min(c0, DM - 1)], 0.f);
    const float v1 = fmaxf(acc1[r] + bb[min(c1, DM - 1)], 0.f);
    O[(size_t)row * DMP + c1] = (c1 < DM) ? v1 : 0.f;
  }
}

// ---------------------------------------------------------------------------
// K7: attention per (b,n,h,tq) — tiny 12x12x15, VALU; reads padded qkv,
// writes o in dense (B,T,N,D) layout
// ---------------------------------------------------------------------------
__global__ __launch_bounds__(256)
void k_attention(const float* __restrict__ q, const float* __restrict__ k,
                 const float* __restrict__ v, float* __restrict__ o) {
  const int tid = blockIdx.x * 256 + threadIdx.x;
  const int tq = tid % SEQT;
  const int h = (tid / SEQT) % NH;
  const int n = (tid / (SEQT * NH)) % NN;
  const int b = tid / (SEQT * NH * NN);
  float qr[HD];
  {
    const float* qp = q + ((size_t)(b * SEQT + tq) * NNP + n) * DMP + h * HD;
    for (int e = 0; e < HD; ++e) qr[e] = qp[e];
  }
  const float scale = 0.25819888974716112f;  // 1/sqrt(15)
  float s[SEQT], mx = -1e30f;
  for (int tk = 0; tk < SEQT; ++tk) {
    const float* kp = k + ((size_t)(b * SEQT + tk) * NNP + n) * DMP + h * HD;
    float acc = 0.f;
    for (int e = 0; e < HD; ++e) acc += qr[e] * kp[e];
    s[tk] = acc * scale;
    mx = fmaxf(mx, s[tk]);
  }
  float sum = 0.f;
  for (int tk = 0; tk < SEQT; ++tk) { s[tk] = expf(s[tk] - mx); sum += s[tk]; }
  const float inv = 1.f / sum;
  float ov[HD];
  for (int e = 0; e < HD; ++e) ov[e] = 0.f;
  for (int tk = 0; tk < SEQT; ++tk) {
    const float* vp = v + ((size_t)(b * SEQT + tk) * NNP + n) * DMP + h * HD;
    const float w = s[tk] * inv;
    for (int e = 0; e < HD; ++e) ov[e] += w * vp[e];
  }
  float* op = o + ((size_t)(b * SEQT + tq) * NN + n) * DM + h * HD;
  for (int e = 0; e < HD; ++e) op[e] = ov[e];
}

// ---------------------------------------------------------------------------
// K8: P = o @ out_w + out_b (65280x120x120); 2x2 register blocking:
// 4 wmma per 4 b64 loads. Writes padded pbuf (stride DMP).
// ---------------------------------------------------------------------------
__global__ __launch_bounds__(256)
void k_proj(const float* __restrict__ o, const float* __restrict__ owt,
            const float* __restrict__ bias, float* __restrict__ out) {
  const int gid = blockIdx.x * 8 + (threadIdx.x >> 5);
  const int mt2 = gid / 4, np = gid % 4;
  if (mt2 >= ROWS / 32) return;
  const int lane = threadIdx.x & 31, ln = lane & 15, hi = lane >> 4;
  const float* A0 = o + (size_t)(mt2 * 32 + ln) * DM;
  const float* A1 = A0 + (size_t)16 * DM;
  const int n0 = np * 32 + ln;
  const float* B0 = owt + (size_t)n0 * DMP;
  const float* B1 = B0 + (size_t)16 * DMP;
  v8f acc00 = {}, acc01 = {}, acc10 = {}, acc11 = {};
  for (int k0 = 0; k0 < DM; k0 += 4) {
    const int ka = k0 + 2 * hi;
    const v2f a0 = ld2(A0 + ka), a1 = ld2(A1 + ka);
    const v2f b0 = ld2(B0 + ka), b1 = ld2(B1 + ka);
    acc00 = wmma4(a0, b0, acc00);
    acc01 = wmma4(a0, b1, acc01);
    acc10 = wmma4(a1, b0, acc10);
    acc11 = wmma4(a1, b1, acc11);
  }
  const int c0 = np * 32 + ln, c1 = c0 + 16;
  const float bb0 = bias[min(c0, DM - 1)], bb1 = bias[min(c1, DM - 1)];
  for (int r = 0; r < 8; ++r) {
    const int row0 = mt2 * 32 + r + 8 * hi, row1 = row0 + 16;
    out[(size_t)row0 * DMP + c0] = acc00[r] + bb0;
    out[(size_t)row0 * DMP + c1] = acc01[r] + bb1;
    out[(size_t)row1 * DMP + c0] = acc10[r] + bb0;
    out[(size_t)row1 * DMP + c1] = acc11[r] + bb1;
  }
}

// ---------------------------------------------------------------------------
// K9: out1 = LayerNorm(x + P) (1 wave / 120-wide row; clamp+select loads)
// ---------------------------------------------------------------------------
__global__ __launch_bounds__(256)
void k_ln1(const float* __restrict__ x, const float* __restrict__ p,
           const float* __restrict__ g, const float* __restrict__ b,
           float* __restrict__ out) {
  const int row = blockIdx.x * 8 + (threadIdx.x >> 5);
  if (row >= ROWS) return;
  const int lane = threadIdx.x & 31;
  const float* xp = x + (size_t)row * DM;
  const float* pp = p + (size_t)row * DMP;
  float vals[4], s = 0.f;
  for (int i = 0; i < 4; ++i) {
    const int j = lane + i * 32;
    const int jc = min(j, DM - 1);
    const float t = xp[jc] + pp[jc];
    vals[i] = (j < DM) ? t : 0.f;
    s += vals[i];
  }
  const float mean = wave_sum(s) / DM;
  float vs = 0.f;
  for (int i = 0; i < 4; ++i) {
    const int j = lane + i * 32;
    const float dv = (j < DM) ? vals[i] - mean : 0.f;
    vs += dv * dv;
  }
  const float rstd = rsqrtf(wave_sum(vs) / DM + LEPS);
  float* op = out + (size_t)row * DM;
  for (int i = 0; i < 4; ++i) {
    const int j = lane + i * 32;
    if (j < DM) op[j] = (vals[i] - mean) * rstd * g[j] + b[j];
  }
}

// ---------------------------------------------------------------------------
// K10: fused FFN + residual + LayerNorm2 per 16-row tile; hidden (16x2048)
// stays in LDS (4 chunks of 512). A-tile staged via async global->LDS when
// the toolchain exposes it; streamed weight rows prefetched.
// ---------------------------------------------------------------------------
#define CHUNK 512
__global__ __launch_bounds__(256, 1)
void k_ffn_ln2(const float* __restrict__ in, const float* __restrict__ w1t,
               const float* __restrict__ b1, const float* __restrict__ w2t,
               const float* __restrict__ b2, const float* __restrict__ g,
               const float* __restrict__ bt, float* __restrict__ out) {
  __shared__ float Asm[16][132];
  __shared__ float Hsm[16][CHUNK + 4];
  __shared__ float Osm[16][132];
  const int r0 = blockIdx.x * 16;
  const int tid = threadIdx.x;
#if defined(HAVE_ASYNC_LDS)
  // async stage: 16 rows x 120 floats = 480 b128 transfers (pad cols never read)
  {
    __attribute__((address_space(3))) float* lbase =
        (__attribute__((address_space(3))) float*)&Asm[0][0];
    for (int idx = tid; idx < 16 * 30; idx += 256) {
      const int row = idx / 30, c4 = (idx % 30) * 4;
      __builtin_amdgcn_global_load_async_to_lds_b128(
          (__attribute__((address_space(1))) v4i*)(void*)(in + (size_t)(r0 + row) * DM + c4),
          (__attribute__((address_space(3))) v4i*)(lbase + row * 132 + c4),
          0, 0);
    }
    __builtin_amdgcn_s_wait_asynccnt(0);
  }
#else
  for (int idx = tid; idx < 16 * 128; idx += 256) {
    const int row = idx >> 7, col = idx & 127;
    const float t = in[(size_t)(r0 + row) * DM + min(col, DM - 1)];
    Asm[row][col] = (col < DM) ? t : 0.f;
  }
#endif
  __syncthreads();
  const int wid = tid >> 5, lane = tid & 31, ln = lane & 15, hi = lane >> 4;
  const int col2 = wid * 16 + ln;
  const float* B2row = w2t + (size_t)col2 * FF;   // pad rows (>=120) are zero
  v8f acc2 = {};
  for (int c = 0; c < FF / CHUNK; ++c) {
    // GEMM1 partial: H[:, c*512..) = relu(A @ W1 + b1); 2 tiles per wave pass
    for (int ntp = wid; ntp < CHUNK / 32; ntp += 8) {
      const int cg0 = c * CHUNK + ntp * 32 + ln;
      const float* B0 = w1t + (size_t)cg0 * DMP;
      const float* B1 = B0 + (size_t)16 * DMP;
      if (ntp + 8 < CHUNK / 32) {     // prefetch next pair's rows (this wave)
        __builtin_prefetch(B0 + (size_t)(8 * 32) * DMP, 0, 3);
        __builtin_prefetch(B1 + (size_t)(8 * 32) * DMP, 0, 3);
      }
      v8f a0 = {}, a1 = {};
      for (int k0 = 0; k0 < DM; k0 += 4) {
        const int ka = k0 + 2 * hi;
        const v2f a = ld2(&Asm[ln][ka]);
        a0 = wmma4(a, ld2(B0 + ka), a0);
        a1 = wmma4(a, ld2(B1 + ka), a1);
      }
      const float bb0 = b1[cg0], bb1 = b1[cg0 + 16];
      for (int r = 0; r < 8; ++r) {
        Hsm[r + 8 * hi][ntp * 32 + ln] = fmaxf(a0[r] + bb0, 0.f);
        Hsm[r + 8 * hi][ntp * 32 + 16 + ln] = fmaxf(a1[r] + bb1, 0.f);
      }
    }
    __syncthreads();
    // prefetch next chunk of this wave's w2t row (2KB; 128B per lane)
    if (c + 1 < FF / CHUNK)
      __builtin_prefetch(B2row + (size_t)(c + 1) * CHUNK + lane * 32, 0, 3);
    // GEMM2 partial: acc2 += H_chunk @ W2[c*512.., :]
    for (int k0 = 0; k0 < CHUNK; k0 += 4) {
      const int ka = k0 + 2 * hi;
      acc2 = wmma4(ld2(&Hsm[ln][ka]), ld2(B2row + c * CHUNK + ka), acc2);
    }
    __syncthreads();
  }
  {
    const float bb = b2[min(col2, DM - 1)];
    for (int r = 0; r < 8; ++r) {
      const int row = r + 8 * hi;
      const float v = acc2[r] + bb + Asm[row][col2];
      Osm[row][col2] = (col2 < DM) ? v : 0.f;
    }
  }
  __syncthreads();
  for (int rr = 0; rr < 2; ++rr) {
    const int row = wid + rr * 8;
    float vals[4], s = 0.f;
    for (int i = 0; i < 4; ++i) {
      const int j = lane + i * 32;
      const float t = Osm[row][min(j, DM - 1)];
      vals[i] = (j < DM) ? t : 0.f;
      s += vals[i];
    }
    const float mean = wave_sum(s) / DM;
    float vs = 0.f;
    for (int i = 0; i < 4; ++i) {
      const int j = lane + i * 32;
      const float dv = (j < DM) ? vals[i] - mean : 0.f;
      vs += dv * dv;
    }
    const float rstd = rsqrtf(wave_sum(vs) / DM + LEPS);
    float* op = out + (size_t)(r0 + row) * DM;
    for (int i = 0; i < 4; ++i) {
      const int j = lane + i * 32;
      if (j < DM) op[j] = (vals[i] - mean) * rstd * g[j] + bt[j];
    }
  }
}

// ---------------------------------------------------------------------------
extern "C" void kernel_launch(void* const* d_in, const int* in_sizes, int n_in,
                              void* d_out, int out_size, void* d_ws, size_t ws_size,
                              hipStream_t stream) {
  (void)in_sizes; (void)n_in; (void)out_size; (void)ws_size;
  const float* x       = (const float*)d_in[0];
  const float* adj     = (const float*)d_in[1];
  const float* gl_beta = (const float*)d_in[2];
  const float* gl_w1   = (const float*)d_in[3];
  const float* gl_w2   = (const float*)d_in[4];
  const float* gl_cw   = (const float*)d_in[5];
  const float* gl_cb   = (const float*)d_in[6];
  const float* cheb_w  = (const float*)d_in[7];
  const float* cheb_b  = (const float*)d_in[8];
  const float* out_w   = (const float*)d_in[9];
  const float* out_b   = (const float*)d_in[10];
  const float* ff_w1   = (const float*)d_in[11];
  const float* ff_b1   = (const float*)d_in[12];
  const float* ff_w2   = (const float*)d_in[13];
  const float* ff_b2   = (const float*)d_in[14];
  const float* ln1_g   = (const float*)d_in[15];
  const float* ln1_b   = (const float*)d_in[16];
  const float* ln2_g   = (const float*)d_in[17];
  const float* ln2_b   = (const float*)d_in[18];
  float* out = (float*)d_out;
  float* ws  = (float*)d_ws;

  // workspace layout (floats), ~53.1M total
  float* bufA  = ws;                                   // 3*NNP2
  float* bufB  = bufA  + (size_t)3 * NNP2;
  float* Lbuf  = bufB  + (size_t)3 * NNP2;
  float* Ltbuf = Lbuf  + (size_t)3 * NNP2;
  float* T2buf = Ltbuf + (size_t)3 * NNP2;
  float* dbuf  = T2buf + (size_t)3 * NNP2;             // 3*256
  float* cwt   = dbuf  + 3 * 256;                      // 9*128*128
  float* owt   = cwt   + (size_t)9 * DMP * DMP;        // 128*128
  float* w1t   = owt   + (size_t)DMP * DMP;            // 2048*128
  float* w2t   = w1t   + (size_t)FF * DMP;             // 128*2048
  float* xt    = w2t   + (size_t)DMP * FF;             // PBSZ
  float* qkv   = xt    + PBSZ;                         // 3*PBSZ
  float* Zbuf  = qkv   + 3 * PBSZ;                     // 2*PBSZ
  float* obuf  = Zbuf;                                 // alias: dense 65280x120
  float* pbuf  = Zbuf + PBSZ;                          // alias: padded 65280x128
  float* out1  = qkv;                                  // alias: dense 65280x120

  // --- prep: transposed+padded weights, transposed x ---
  k_transpose_pad<<<dim3(64, 9), 256, 0, stream>>>(cheb_w, cwt, DM, DM, DMP, DMP);
  k_transpose_pad<<<dim3(64, 1), 256, 0, stream>>>(out_w, owt, DM, DM, DMP, DMP);
  k_transpose_pad<<<dim3(1024, 1), 256, 0, stream>>>(ff_w1, w1t, DM, FF, DMP, FF);
  k_transpose_pad<<<dim3(1024, 1), 256, 0, stream>>>(ff_w2, w2t, FF, DM, FF, DMP);
  k_build_xt<<<(int)(PBSZ / 256), 256, 0, stream>>>(x, xt);

  // --- graph learn (3 branches) ---
  k_graph_new<<<dim3(NTI * NTI, 3), 32, 0, stream>>>(adj, gl_w1, gl_w2, gl_beta,
                                                     gl_cw, gl_cb, bufA);
  k_rowsum_rsqrt<<<dim3(NNP, 3), 32, 0, stream>>>(bufA, dbuf);
  k_sym_norm<<<dim3(NNP2 / 256, 3), 256, 0, stream>>>(bufA, dbuf, bufB, 0, 0.5f / NN);
  k_rowsum_rsqrt<<<dim3(NNP, 3), 32, 0, stream>>>(bufB, dbuf);
  k_sym_norm<<<dim3(NNP2 / 256, 3), 256, 0, stream>>>(bufB, dbuf, bufA, 1, 0.f);

  // --- Chebyshev basis: L = I - d g d; Lt; T2 = 2 L^2 - I ---
  k_rowsum_rsqrt<<<dim3(NNP, 3), 32, 0, stream>>>(bufA, dbuf);
  k_sym_norm<<<dim3(NNP2 / 256, 3), 256, 0, stream>>>(bufA, dbuf, Lbuf, 2, 0.f);
  k_transpose_g<<<dim3(NNP2 / 256, 3), 256, 0, stream>>>(Lbuf, Ltbuf);
  k_cheb_t2<<<dim3(NTI * NTI, 3), 32, 0, stream>>>(Lbuf, Ltbuf, T2buf);

  // --- Cheb conv per branch (Z reused -> sequential branches) ---
  for (int br = 0; br < 3; ++br) {
    k_cheb_z<<<(BT * 88) / 8, 256, 0, stream>>>(Lbuf, T2buf, xt, Zbuf, br);
    k_cheb_y<<<(BT * 44) / 8, 256, 0, stream>>>(x, Zbuf, cwt, cheb_b,
                                                qkv + (size_t)br * PBSZ, br);
  }

  // --- attention + projection + LN1 ---
  k_attention<<<(BB * NN * NH * SEQT) / 256, 256, 0, stream>>>(
      qkv, qkv + PBSZ, qkv + 2 * PBSZ, obuf);
  k_proj<<<(ROWS / 32 * 4) / 8, 256, 0, stream>>>(obuf, owt, out_b, pbuf);
  k_ln1<<<ROWS / 8, 256, 0, stream>>>(x, pbuf, ln1_g, ln1_b, out1);

  // --- fused FFN + residual + LN2 ---
  k_ffn_ln2<<<ROWS / 16, 256, 0, stream>>>(out1, w1t, ff_b1, w2t, ff_b2,
                                           ln2_g, ln2_b, out);
}